// ConvolutionalCapsule_76759655514980
// MI455X (gfx1250) — compile-verified
//
#include <hip/hip_runtime.h>
#include <math.h>

// CDNA5 / gfx1250 fused convolutional-capsule + EM-routing kernel.
// One workgroup (256 threads = 8 wave32) per output position (288 total).
// Votes V computed with v_wmma_f32_16x16x4_f32, kept resident in LDS as bf16.

typedef __attribute__((ext_vector_type(2))) float v2f;
typedef __attribute__((ext_vector_type(8))) float v8f;

#define EPSF 1e-7f

// LDS carve (bytes):
//   V (bf16)          : 288*32*16*2 = 294912
//   aI (f32)          : 288*4       = 1152
//   S1,S2 (f32)       : 2*512*4     = 4096
//   red (f32)         : 32*17*4     = 2176
//   Mj,I2V (f32)      : 2*512*4     = 4096
//   Aj,LgA (f32)      : 2*32*4      = 256
#define SMEM_BYTES (294912 + 1152 + 4096 + 2176 + 4096 + 256)

__device__ __forceinline__ unsigned short f32_to_bf16(float f) {
  unsigned u = __float_as_uint(f);
  u += 0x7FFFu + ((u >> 16) & 1u);            // round-to-nearest-even
  return (unsigned short)(u >> 16);
}
__device__ __forceinline__ float bf16_to_f32(unsigned h) {
  return __uint_as_float(h << 16);
}
__device__ __forceinline__ void unpack8(uint4 u, float* v) {
  v[0] = bf16_to_f32(u.x & 0xFFFFu); v[1] = bf16_to_f32(u.x >> 16);
  v[2] = bf16_to_f32(u.y & 0xFFFFu); v[3] = bf16_to_f32(u.y >> 16);
  v[4] = bf16_to_f32(u.z & 0xFFFFu); v[5] = bf16_to_f32(u.z >> 16);
  v[6] = bf16_to_f32(u.w & 0xFFFFu); v[7] = bf16_to_f32(u.w >> 16);
}

// ---------------------------------------------------------------------------
// Kernel 0: stage W (288,32,4,4) into the WMMA B-operand lane layout so the
// main kernel can load its per-lane v2f B operand with one contiguous b64.
// B tile for capsule i, N-tile j: lane n (n<16): N=j*16+n, K-half 0 (q=0,1);
// lane n (n>=16): N=j*16+(n-16), K-half 1 (q=2,3). VGPR0=first q of half,
// VGPR1=second q of half. N index encodes (c,r) as cr=c*4+r.
// ---------------------------------------------------------------------------
__global__ __launch_bounds__(256) void stage_w_kernel(const float* __restrict__ W,
                                                      float* __restrict__ S) {
  int idx  = blockIdx.x * blockDim.x + threadIdx.x;   // 288*8*32 = 73728
  int i    = idx >> 8;
  int rem  = idx & 255;
  int j    = rem >> 5;
  int lane = rem & 31;
  int cr   = j * 16 + (lane & 15);
  int c    = cr >> 2;
  int r    = cr & 3;
  int qb   = (lane >> 4) * 2;
  const float* wp = W + i * 512 + c * 16 + r;   // W[i][c][q][r], stride over q = 4
  S[idx * 2 + 0] = wp[qb * 4];
  S[idx * 2 + 1] = wp[(qb + 1) * 4];
}

// ---------------------------------------------------------------------------
// Main fused kernel.
// ---------------------------------------------------------------------------
__global__ __launch_bounds__(256, 1) void capsule_fused_kernel(
    const float* __restrict__ x,      // (2,14,14,32,17)
    const float* __restrict__ Wstg,   // staged W
    const float* __restrict__ beta_v, // (32)
    const float* __restrict__ beta_a, // (32)
    float* __restrict__ out)          // (2,12,12,32,17)
{
  extern __shared__ char smem[];
  unsigned short* Vlds = (unsigned short*)smem;              // 147456 bf16
  float* aI  = (float*)(smem + 294912);                      // [288]
  float* S1  = aI + 288;                                     // [32][16] sum V
  float* S2  = S1 + 512;                                     // [32][16] sum V^2
  float* red = S2 + 512;                                     // [32][17] Rsum + sum Rw*V
  float* Mj  = red + 32 * 17;                                // [32][16]
  float* I2V = Mj + 512;                                     // [32][16] 1/(2*sigma^2)
  float* Aj  = I2V + 512;                                    // [32]
  float* LgA = Aj + 32;                                      // [32]

  const int tid  = threadIdx.x;
  const int lane = tid & 31;     // = output capsule c in routing phases
  const int w    = tid >> 5;     // wave id 0..7
  const int pos  = blockIdx.x;   // 0..287
  const int b    = pos / 144;
  const int hw   = pos % 144;
  const int ho   = hw / 12;
  const int wo   = hw % 12;

  // ---- init: zero S1/S2, load patch activations ---------------------------
  for (int t = tid; t < 1024; t += 256) S1[t] = 0.0f;   // S1 and S2 contiguous
  for (int i = tid; i < 288; i += 256) {
    int ki = i / 96, kj = (i % 96) >> 5, cin = i & 31;
    aI[i] = x[((((b * 14) + (ho + ki)) * 14 + (wo + kj)) * 32 + cin) * 17 + 16];
  }

  // ---- vote transform via v_wmma_f32_16x16x4_f32 --------------------------
  // A (16x4 f32): lanes 0-15 rows M=0..15 K-half {0,1}; lanes 16-31 K-half {2,3}.
  // Rows 0..3 = pose rows p; rows 4..15 zero (D rows 4..15 unused).
  // Per capsule: issue all 8 B-tile loads first (one load clause in flight),
  // then run the 8 WMMAs back-to-back, then a single predicated store phase.
  {
    const int rowA  = lane & 15;
    const int halfA = lane >> 4;
    for (int i = w; i < 288; i += 8) {
      int ki = i / 96, kj = (i % 96) >> 5, cin = i & 31;
      const float* xp = x + ((((b * 14) + (ho + ki)) * 14 + (wo + kj)) * 32 + cin) * 17;
      v2f A; A.x = 0.0f; A.y = 0.0f;
      if (rowA < 4) {
        int e = rowA * 4 + halfA * 2;   // pose element p*4+q
        A.x = xp[e];
        A.y = xp[e + 1];
      }
      const float* bp = Wstg + (((i * 8) * 32) + lane) * 2;
      if (i + 8 < 288)   // pull next capsule's staged-W block toward the WGP
        __builtin_prefetch(Wstg + ((((i + 8) * 8) * 32) + lane) * 2, 0, 3);

      // 1) all B operands in flight
      v2f Bt[8];
      #pragma unroll
      for (int j = 0; j < 8; ++j) {
        Bt[j].x = bp[j * 64 + 0];
        Bt[j].y = bp[j * 64 + 1];
      }
      // 2) dense WMMA chain
      v8f D[8];
      #pragma unroll
      for (int j = 0; j < 8; ++j) {
        v8f C = {};
        D[j] = __builtin_amdgcn_wmma_f32_16x16x4_f32(
            /*neg_a=*/false, A, /*neg_b=*/false, Bt[j],
            /*c_mod=*/(short)0, C, /*reuse_a=*/false, /*reuse_b=*/false);
      }
      // 3) single predicated pack/store phase
      if (lane < 16) {
        #pragma unroll
        for (int j = 0; j < 8; ++j) {
          // D VGPR k (k=0..3) on lanes 0-15 = V[p=k][n=lane] for this N-tile.
          int cr = j * 16 + lane;                        // cr = c*4 + r
          unsigned lo = (unsigned)f32_to_bf16(D[j][0]) | ((unsigned)f32_to_bf16(D[j][1]) << 16);
          unsigned hi = (unsigned)f32_to_bf16(D[j][2]) | ((unsigned)f32_to_bf16(D[j][3]) << 16);
          uint2 pk; pk.x = lo; pk.y = hi;
          // layout: Vlds[i*512 + cr*4 + p]  (within-c index k_lds = r*4+p)
          *(uint2*)(Vlds + i * 512 + cr * 4) = pk;
        }
      }
    }
  }
  __syncthreads();

  // ---- one-time unweighted sums over i: S1=sum V, S2=sum V^2 (per c,k) ----
  {
    float s1[16], s2[16];
    #pragma unroll
    for (int k = 0; k < 16; ++k) { s1[k] = 0.0f; s2[k] = 0.0f; }
    #pragma unroll
    for (int n = 0; n < 36; ++n) {
      int i = w + n * 8;
      const uint4* vp = (const uint4*)(Vlds + i * 512 + lane * 16);
      uint4 ua = vp[0], ub = vp[1];
      float v[16];
      unpack8(ua, v); unpack8(ub, v + 8);
      #pragma unroll
      for (int k = 0; k < 16; ++k) {
        s1[k] += v[k];
        s2[k] = fmaf(v[k], v[k], s2[k]);
      }
    }
    #pragma unroll
    for (int k = 0; k < 16; ++k) {
      atomicAdd(&S1[lane * 16 + k], s1[k]);
      atomicAdd(&S2[lane * 16 + k], s2[k]);
    }
  }
  __syncthreads();

  // ---- EM routing: 3 iterations ------------------------------------------
  float R[36];
  #pragma unroll
  for (int n = 0; n < 36; ++n) R[n] = 1.0f / 32.0f;

  for (int it = 0; it < 3; ++it) {
    for (int t = tid; t < 32 * 17; t += 256) red[t] = 0.0f;
    __syncthreads();

    // per-thread weighted partials: Rsum, sum Rw*V   (c = lane)
    {
      float rs = 0.0f, ws[16];
      #pragma unroll
      for (int k = 0; k < 16; ++k) ws[k] = 0.0f;
      #pragma unroll
      for (int n = 0; n < 36; ++n) {
        int i = w + n * 8;
        float rw = R[n] * aI[i];
        rs += rw;
        const uint4* vp = (const uint4*)(Vlds + i * 512 + lane * 16);
        uint4 ua = vp[0], ub = vp[1];
        float v[16];
        unpack8(ua, v); unpack8(ub, v + 8);
        #pragma unroll
        for (int k = 0; k < 16; ++k) ws[k] = fmaf(rw, v[k], ws[k]);
      }
      atomicAdd(&red[lane * 17], rs);
      #pragma unroll
      for (int k = 0; k < 16; ++k) atomicAdd(&red[lane * 17 + 1 + k], ws[k]);
    }
    __syncthreads();

    // wave 0: per-c M-step + activation (lane = c), cross-c stats via shfl
    if (tid < 32) {
      int c = tid;
      float rsum = red[c * 17];
      float invr = 1.0f / rsum;
      float sumlog = 0.0f;
      #pragma unroll
      for (int k = 0; k < 16; ++k) {
        float M = red[c * 17 + 1 + k] * invr;
        Mj[c * 16 + k] = M;
        // reference stdv: sqrt( sum_i (V - M)^2 )  (unweighted; Rsum cancels)
        float vs = S2[c * 16 + k] - 2.0f * M * S1[c * 16 + k] + 288.0f * M * M;
        float sd = sqrtf(fmaxf(vs, 0.0f));
        sumlog += logf(sd + EPSF);
        I2V[c * 16 + k] = 0.5f / (sd * sd + 1e-30f);
      }
      float costv = rsum * (16.0f * beta_v[c] + sumlog);
      float m = costv;
      #pragma unroll
      for (int o = 16; o; o >>= 1) m += __shfl_xor(m, o, 32);
      m *= (1.0f / 32.0f);
      float d = costv - m;
      float vv = d * d;
      #pragma unroll
      for (int o = 16; o; o >>= 1) vv += __shfl_xor(vv, o, 32);
      float cstd = sqrtf(vv * (1.0f / 32.0f));
      float inv_temp = 1.0f + (float)it;     // 1 + (3-1)*it/2
      float ac = beta_a[c] + (m - costv) / (cstd + EPSF);
      float ajv = 1.0f / (1.0f + expf(-inv_temp * ac));
      Aj[c]  = ajv;
      LgA[c] = logf(ajv + EPSF) - sumlog;
    }
    __syncthreads();

    // E-step: update R with softmax over c (one c per lane -> wave32 shuffles)
    if (it < 2) {
      float lM[16], lI[16];
      #pragma unroll
      for (int k = 0; k < 16; ++k) { lM[k] = Mj[lane * 16 + k]; lI[k] = I2V[lane * 16 + k]; }
      float la = LgA[lane];
      #pragma unroll
      for (int n = 0; n < 36; ++n) {
        int i = w + n * 8;
        const uint4* vp = (const uint4*)(Vlds + i * 512 + lane * 16);
        uint4 ua = vp[0], ub = vp[1];
        float v[16];
        unpack8(ua, v); unpack8(ub, v + 8);
        float s = 0.0f;
        #pragma unroll
        for (int k = 0; k < 16; ++k) {
          float dv = v[k] - lM[k];
          s = fmaf(dv * dv, lI[k], s);
        }
        float lp = la - s;
        float mx = lp;
        #pragma unroll
        for (int o = 16; o; o >>= 1) mx = fmaxf(mx, __shfl_xor(mx, o, 32));
        float e = expf(lp - mx);
        float se = e;
        #pragma unroll
        for (int o = 16; o; o >>= 1) se += __shfl_xor(se, o, 32);
        R[n] = e / se;
      }
      __syncthreads();
    }
  }

  // ---- write output: (pos, c, 17): pose (p*4+r order) + sigmoid(a_j) ------
  if (tid < 32) {
    int c = tid;
    float* op = out + (pos * 32 + c) * 17;
    #pragma unroll
    for (int p = 0; p < 4; ++p)
      #pragma unroll
      for (int r = 0; r < 4; ++r)
        op[p * 4 + r] = Mj[c * 16 + r * 4 + p];   // LDS holds k_lds = r*4+p
    op[16] = 1.0f / (1.0f + expf(-Aj[c]));
  }
}

// ---------------------------------------------------------------------------
extern "C" void kernel_launch(void* const* d_in, const int* in_sizes, int n_in,
                              void* d_out, int out_size, void* d_ws, size_t ws_size,
                              hipStream_t stream) {
  (void)in_sizes; (void)n_in; (void)out_size; (void)ws_size;
  const float* x  = (const float*)d_in[0];   // (2,14,14,32,17)
  const float* W  = (const float*)d_in[1];   // (288,32,4,4)
  const float* bv = (const float*)d_in[2];   // (32)
  const float* ba = (const float*)d_in[3];   // (32)
  float* out = (float*)d_out;
  float* stg = (float*)d_ws;                 // 589824 bytes used

  static_assert(SMEM_BYTES <= 320 * 1024, "exceeds WGP LDS");
  hipFuncSetAttribute(reinterpret_cast<const void*>(capsule_fused_kernel),
                      hipFuncAttributeMaxDynamicSharedMemorySize, SMEM_BYTES);

  stage_w_kernel<<<288, 256, 0, stream>>>(W, stg);
  capsule_fused_kernel<<<288, 256, SMEM_BYTES, stream>>>(x, stg, bv, ba, out);
}